// LinearSequenceToSequenceClassificationFromAttentionHead_83013127897680
// MI455X (gfx1250) — compile-verified
//
#include <hip/hip_runtime.h>
#include <hip/hip_bf16.h>
#include <float.h>

typedef __attribute__((ext_vector_type(2))) float v2f;
typedef __attribute__((ext_vector_type(8))) float v8f;

#define B_  8
#define H_  16
#define S_  1024
#define K_  10
#define L_  8
#define F_  (H_ * K_)   // 160 features per (b, s)

// ---------------------------------------------------------------------------
// Kernel 1: streaming top-10 over attention columns.
// One lane per column s of one (b,h) slice; wave reads 32 consecutive floats
// per row -> fully coalesced 128B transactions. Non-temporal loads (data is
// single-use) + prefetch hint a few rows ahead.
// feats layout: [(b*S + s)*160 + h*10 + k]  (contiguous 160-float rows).
// ---------------------------------------------------------------------------
__global__ __launch_bounds__(256) void topk_cols_kernel(
    const float* __restrict__ att, float* __restrict__ feats) {
  const int s = blockIdx.x * 256 + threadIdx.x;
  const int h = blockIdx.y;
  const int b = blockIdx.z;

  const float* __restrict__ p =
      att + ((size_t)(b * H_ + h) * S_) * S_ + (size_t)s;

  float t[K_];
#pragma unroll
  for (int i = 0; i < K_; ++i) t[i] = -FLT_MAX;

  auto insert = [&](float v) {
    if (v > t[K_ - 1]) {            // rare path (~5% of elements)
#pragma unroll
      for (int i = 0; i < K_; ++i) {
        float hi = fmaxf(t[i], v);
        v        = fminf(t[i], v);
        t[i]     = hi;
      }
    }
  };

  for (int r = 0; r < S_; r += 4) {
    if (r + 16 < S_) __builtin_prefetch(p + (size_t)(r + 16) * S_, 0, 0);
    float v0 = __builtin_nontemporal_load(p + (size_t)(r + 0) * S_);
    float v1 = __builtin_nontemporal_load(p + (size_t)(r + 1) * S_);
    float v2 = __builtin_nontemporal_load(p + (size_t)(r + 2) * S_);
    float v3 = __builtin_nontemporal_load(p + (size_t)(r + 3) * S_);
    insert(v0); insert(v1); insert(v2); insert(v3);
  }

  float* __restrict__ o = feats + (size_t)(b * S_ + s) * F_ + h * K_;
#pragma unroll
  for (int i = 0; i < K_; ++i) o[i] = t[i];
}

// ---------------------------------------------------------------------------
// Kernel 2: logits = feats(8192x160) @ W^T(160x8) + b via V_WMMA_F32_16X16X4_F32.
// Each wave owns a 16-row tile of positions and the full (padded) 16-label
// column block; K=160 is consumed in 40 chained fp32 WMMAs (exact fp32 math).
// A 16x4 layout: lanes 0-15 -> M=lane, K = k0+{0,1}; lanes 16-31 -> K = k0+{2,3}.
// B 4x16 mirrors A from W (labels 8..15 zero-padded with cndmask, EXEC stays
// all-ones across the WMMA as required).
// ---------------------------------------------------------------------------
__global__ __launch_bounds__(256) void linear_wmma_kernel(
    const float* __restrict__ feats, const float* __restrict__ W,
    const float* __restrict__ bias, float* __restrict__ out) {
  const int lane    = threadIdx.x & 31;
  const int wave    = blockIdx.x * (blockDim.x >> 5) + (threadIdx.x >> 5);
  const int rowBase = wave * 16;          // 16 (b,s) positions per wave
  const int half    = lane >> 4;          // 0: lanes 0-15, 1: lanes 16-31
  const int lid     = lane & 15;
  const int n       = lid;                // label column (0..15, valid < 8)
  const bool nvalid = (n < L_);

  const float* __restrict__ arow =
      feats + (size_t)(rowBase + lid) * F_ + half * 2;
  const float* __restrict__ wrow =
      W + (size_t)(nvalid ? n : 0) * F_ + half * 2;

  v8f acc = {};
  for (int k0 = 0; k0 < F_; k0 += 4) {
    v2f a = *(const v2f*)(arow + k0);
    v2f w = *(const v2f*)(wrow + k0);
    v2f bb;
    bb.x = nvalid ? w.x : 0.0f;           // cndmask, no EXEC divergence
    bb.y = nvalid ? w.y : 0.0f;
    acc = __builtin_amdgcn_wmma_f32_16x16x4_f32(
        /*neg_a=*/false, a, /*neg_b=*/false, bb,
        /*c_mod=*/(short)0, acc, /*reuse_a=*/false, /*reuse_b=*/false);
  }

  if (nvalid) {
    const float bv = bias[n];
#pragma unroll
    for (int v = 0; v < 8; ++v) {
      const int m = v + half * 8;         // D: VGPR v -> row v / v+8
      out[(size_t)(rowBase + m) * L_ + n] = acc[v] + bv;
    }
  }
}

// ---------------------------------------------------------------------------
extern "C" void kernel_launch(void* const* d_in, const int* in_sizes, int n_in,
                              void* d_out, int out_size, void* d_ws,
                              size_t ws_size, hipStream_t stream) {
  const float* att  = (const float*)d_in[0];  // (8,16,1024,1024) f32
  const float* W    = (const float*)d_in[1];  // (8,160) f32
  const float* bias = (const float*)d_in[2];  // (8,) f32
  float* out        = (float*)d_out;          // (8,1024,8) f32
  float* feats      = (float*)d_ws;           // (8*1024,160) f32 scratch

  // Kernel 1: 4 column-blocks x 16 heads x 8 batches, 256 threads each.
  dim3 grid1(S_ / 256, H_, B_);
  topk_cols_kernel<<<grid1, 256, 0, stream>>>(att, feats);

  // Kernel 2: 8192 rows / 16 per wave = 512 waves = 64 blocks of 8 waves.
  const int totalWaves = (B_ * S_) / 16;               // 512
  const int blocks     = totalWaves / 8;               // 64
  linear_wmma_kernel<<<blocks, 256, 0, stream>>>(feats, W, bias, out);
}